// RoPEAttention_47785806135880
// MI455X (gfx1250) — compile-verified
//
#include <hip/hip_runtime.h>
#include <hip/hip_bf16.h>

// Problem constants from the reference
#define Bb 4
#define Tt 2048
#define Cc 2048
#define NH 16
#define HD 128
#define C3 (3 * Cc)          // 6144
#define MROWS (Bb * Tt)      // 8192
#define KT 32                // GEMM K-tile staged in LDS
#define BCOL_STRIDE 34       // padded LDS column stride (even -> b64 aligned)

typedef __attribute__((ext_vector_type(2))) float v2f;
typedef __attribute__((ext_vector_type(8))) float v8f;

typedef __attribute__((address_space(1))) int gint;   // global int
typedef __attribute__((address_space(3))) int lint;   // LDS int

// CDNA5 fp32 tensor op: D(16x16) = A(16x4) * B(4x16) + C
__device__ __forceinline__ v8f wmma_f32(v2f a, v2f b, v8f c) {
    return __builtin_amdgcn_wmma_f32_16x16x4_f32(
        /*neg_a=*/false, a, /*neg_b=*/false, b,
        /*c_mod=*/(short)0, c, /*reuse_a=*/false, /*reuse_b=*/false);
}

// Async global -> LDS (ASYNCcnt-tracked); fallback: VGPR round trip.
__device__ __forceinline__ void async_ld_b32(const float* g, float* l) {
#if __has_builtin(__builtin_amdgcn_global_load_async_to_lds_b32)
    __builtin_amdgcn_global_load_async_to_lds_b32((gint*)g, (lint*)l, 0, 0);
#else
    *l = *g;
#endif
}

// ---------------------------------------------------------------------------
// GEMM + bias: C[M,N] = A[M,K] * B[K,N] + bias[N]   (all fp32, row-major)
// Block = 256 threads = 8 waves (4 in M, 2 in N); wave tile = 32x32.
// Block tile = 128x64.  B K-tiles (32x64) are async-staged into LDS,
// stored transposed (col-major, stride 34) so compute reads are ds b64.
// Double-buffered: issue async tile kt+1, compute tile kt from LDS.
// Requires M%128==0, N%64==0, K%KT==0.
// ---------------------------------------------------------------------------
__global__ __launch_bounds__(256) void gemm_bias_wmma_f32(
    const float* __restrict__ A, const float* __restrict__ Bm,
    const float* __restrict__ bias, float* __restrict__ Cm,
    int M, int N, int K)
{
    __shared__ float bsh[2][64 * BCOL_STRIDE];   // 2 x 8.5 KB

    const int tid  = threadIdx.x;
    const int lane = tid & 31;
    const int wave = tid >> 5;
    const int wm   = wave & 3;           // wave row (0..3)
    const int wn   = wave >> 2;          // wave col (0..1)
    const int half = lane >> 4;          // 0: lanes 0-15, 1: lanes 16-31
    const int l16  = lane & 15;

    const long m0 = (long)blockIdx.x * 128 + wm * 32;
    const long nb = (long)blockIdx.y * 64;        // block N base
    const int  n0 = wn * 32;                      // wave N offset inside tile

    v8f acc[2][2];
    #pragma unroll
    for (int i = 0; i < 2; ++i)
        #pragma unroll
        for (int j = 0; j < 2; ++j)
            acc[i][j] = (v8f){0.f,0.f,0.f,0.f,0.f,0.f,0.f,0.f};

    // A fragment bases: row (m0 + l16 [+16]), K-offset 2*half
    const float* a0 = A + (m0 + l16) * (long)K + 2 * half;
    const float* a1 = a0 + 16 * (long)K;

    // Cooperative async loader: B K-tile (rows k0..k0+31, cols nb..nb+63)
    // -> LDS transposed: bsh[buf][col*34 + row].  8 b32 per thread,
    // consecutive tid -> consecutive global col (coalesced).
    auto issue_tile = [&](int k0, int buf) {
        #pragma unroll
        for (int i = 0; i < 8; ++i) {
            const int e   = tid + 256 * i;        // 0..2047
            const int row = e >> 6;               // 0..31
            const int col = e & 63;               // 0..63
            async_ld_b32(Bm + (long)(k0 + row) * N + nb + col,
                         &bsh[buf][col * BCOL_STRIDE + row]);
        }
    };

    issue_tile(0, 0);
    asm volatile("s_wait_asynccnt 0" ::: "memory");
    __syncthreads();

    const int NKT = K / KT;
    for (int kt = 0; kt < NKT; ++kt) {
        const int buf = kt & 1;
        if (kt + 1 < NKT) issue_tile((kt + 1) * KT, buf ^ 1);

        const float* ab0 = a0 + kt * KT;
        const float* ab1 = a1 + kt * KT;
        // prefetch A for the tile after next (probe-verified global_prefetch_b8)
        __builtin_prefetch(ab0 + 2 * KT, 0, 0);
        const float* bl = &bsh[buf][(n0 + l16) * BCOL_STRIDE + 2 * half];

        #pragma unroll
        for (int k = 0; k < KT; k += 4) {
            v2f afr0 = *(const v2f*)(ab0 + k);               // global b64
            v2f afr1 = *(const v2f*)(ab1 + k);
            v2f bfr0 = *(const v2f*)(bl + k);                // ds b64
            v2f bfr1 = *(const v2f*)(bl + 16 * BCOL_STRIDE + k);
            acc[0][0] = wmma_f32(afr0, bfr0, acc[0][0]);
            acc[0][1] = wmma_f32(afr0, bfr1, acc[0][1]);
            acc[1][0] = wmma_f32(afr1, bfr0, acc[1][0]);
            acc[1][1] = wmma_f32(afr1, bfr1, acc[1][1]);
        }
        asm volatile("s_wait_asynccnt 0" ::: "memory");
        __syncthreads();
    }

    #pragma unroll
    for (int j = 0; j < 2; ++j) {
        const long col = nb + n0 + 16 * j + l16;
        const float bv = bias[col];
        #pragma unroll
        for (int i = 0; i < 2; ++i) {
            #pragma unroll
            for (int r = 0; r < 8; ++r) {
                const long row = m0 + 16 * i + r + 8 * half;
                Cm[row * (long)N + col] = acc[i][j][r] + bv;
            }
        }
    }
}

// ---------------------------------------------------------------------------
// RoPE in-place on q and k inside the qkv buffer [B, T, 3C].
// One thread per (d, d+64) pair -> race-free in-place rotation.
// ---------------------------------------------------------------------------
__global__ __launch_bounds__(256) void rope_inplace(float* __restrict__ qkv)
{
    long idx = (long)blockIdx.x * blockDim.x + threadIdx.x;
    const int j   = (int)(idx & 63);           // 0..63
    long rest = idx >> 6;
    const int h   = (int)(rest & 15); rest >>= 4;
    const int qk  = (int)(rest & 1);  rest >>= 1;
    const int t   = (int)(rest & (Tt - 1)); rest >>= 11;
    const int b   = (int)rest;

    float* base = qkv + ((long)(b * Tt + t)) * C3 + qk * Cc + h * HD;
    const float x1 = base[j];
    const float x2 = base[j + 64];
    // inv_freq = 10000^(-2j/128) = exp(-j * 2*ln(10000)/128)
    const float inv = __expf(-0.14391156842870208f * (float)j);
    const float ang = (float)t * inv;
    float s, c;
    __sincosf(ang, &s, &c);
    base[j]      = x1 * c - x2 * s;
    base[j + 64] = x2 * c + x1 * s;
}

// ---------------------------------------------------------------------------
// Causal flash attention, fp32 WMMA. One wave per 16-row Q tile.
// Block = 128 threads (4 waves, 4 consecutive q-tiles), grid (T/64, B*NH).
// ---------------------------------------------------------------------------
__global__ __launch_bounds__(128) void flash_attn_wmma_f32(
    const float* __restrict__ qkv, float* __restrict__ yout)
{
    const int lane = threadIdx.x & 31;
    const int wave = threadIdx.x >> 5;
    const int half = lane >> 4;
    const int l16  = lane & 15;

    const int qt = blockIdx.x * 4 + wave;    // q-tile index (0..127)
    const int q0 = qt * 16;
    const int bh = blockIdx.y;
    const int b  = bh >> 4;
    const int h  = bh & 15;

    const long st = C3;                      // row stride in qkv (floats)
    const float* qbase = qkv + ((long)b * Tt) * st + h * HD;
    const float* kbase = qbase + Cc;
    const float* vbase = qbase + 2 * Cc;

    __shared__ float psh[4][16 * 17];        // per-wave padded P tile
    float* pw = psh[wave];

    // Preload Q tile (16x128) as 32 A-fragments
    v2f qf[32];
    {
        const float* qp = qbase + (long)(q0 + l16) * st + 2 * half;
        #pragma unroll
        for (int s = 0; s < 32; ++s)
            qf[s] = *(const v2f*)(qp + 4 * s);
    }

    v8f yacc[8];
    #pragma unroll
    for (int j = 0; j < 8; ++j) yacc[j] = (v8f){0.f,0.f,0.f,0.f,0.f,0.f,0.f,0.f};
    float mrow[8], lrow[8];
    #pragma unroll
    for (int r = 0; r < 8; ++r) { mrow[r] = -3.0e38f; lrow[r] = 0.f; }

    const float scale = 0.08838834764831845f;   // 1/sqrt(128)
    const int nkt = q0 / 16 + 1;                // causal tile count

    for (int kt = 0; kt < nkt; ++kt) {
        const int t0 = kt * 16;
        // ---- S = Q * K^T (16x16), 32 WMMAs over K-dim 128 ----
        v8f S = (v8f){0.f,0.f,0.f,0.f,0.f,0.f,0.f,0.f};
        const float* kp = kbase + (long)(t0 + l16) * st + 2 * half;
        #pragma unroll
        for (int s = 0; s < 32; ++s) {
            v2f kf = *(const v2f*)(kp + 4 * s);  // contiguous -> b64
            S = wmma_f32(qf[s], kf, S);
        }

        // ---- scale + causal mask ----
        const bool diag = (kt == nkt - 1);
        float sv[8];
        #pragma unroll
        for (int r = 0; r < 8; ++r) {
            float v = S[r] * scale;
            const int Mrow = r + 8 * half;
            if (diag && (l16 > Mrow)) v = -3.0e38f;   // col > row on diagonal tile
            sv[r] = v;
        }

        // ---- online softmax (per-row reductions across 16-lane groups) ----
        #pragma unroll
        for (int r = 0; r < 8; ++r) {
            float mx = sv[r];
            mx = fmaxf(mx, __shfl_xor(mx, 1, 32));
            mx = fmaxf(mx, __shfl_xor(mx, 2, 32));
            mx = fmaxf(mx, __shfl_xor(mx, 4, 32));
            mx = fmaxf(mx, __shfl_xor(mx, 8, 32));
            const float mnew  = fmaxf(mrow[r], mx);
            const float alpha = __expf(mrow[r] - mnew);
            const float p     = __expf(sv[r] - mnew);
            float ps = p;
            ps += __shfl_xor(ps, 1, 32);
            ps += __shfl_xor(ps, 2, 32);
            ps += __shfl_xor(ps, 4, 32);
            ps += __shfl_xor(ps, 8, 32);
            lrow[r] = lrow[r] * alpha + ps;
            mrow[r] = mnew;
            sv[r]   = p;
            #pragma unroll
            for (int j = 0; j < 8; ++j) yacc[j][r] *= alpha;
        }

        // ---- stage P (16x16) through LDS to re-layout C-frag -> A-frag ----
        #pragma unroll
        for (int r = 0; r < 8; ++r)
            pw[(r + 8 * half) * 17 + l16] = sv[r];
        asm volatile("s_wait_dscnt 0" ::: "memory");   // order intra-wave LDS RAW

        v2f pa[4];
        #pragma unroll
        for (int s = 0; s < 4; ++s) {
            const int kk = 4 * s + 2 * half;
            pa[s] = (v2f){ pw[l16 * 17 + kk], pw[l16 * 17 + kk + 1] };
        }

        // ---- Y += P * V (16x128), 32 WMMAs ----
        const float* vp = vbase + (long)t0 * st;
        #pragma unroll
        for (int j = 0; j < 8; ++j) {
            #pragma unroll
            for (int s = 0; s < 4; ++s) {
                const int kk = 4 * s + 2 * half;
                const float* vpp = vp + (long)kk * st + 16 * j + l16;
                v2f vf = (v2f){ vpp[0], vpp[st] };
                yacc[j] = wmma_f32(pa[s], vf, yacc[j]);
            }
        }
    }

    // ---- epilogue: normalize and write y[b, q0+M, h*HD + d] ----
    float* yo = yout + ((long)b * Tt + q0) * (long)Cc + h * HD;
    #pragma unroll
    for (int r = 0; r < 8; ++r) {
        const int Mrow = r + 8 * half;
        const float inv = 1.0f / lrow[r];
        #pragma unroll
        for (int j = 0; j < 8; ++j)
            yo[(long)Mrow * Cc + 16 * j + l16] = yacc[j][r] * inv;
    }
}

// ---------------------------------------------------------------------------
extern "C" void kernel_launch(void* const* d_in, const int* in_sizes, int n_in,
                              void* d_out, int out_size, void* d_ws, size_t ws_size,
                              hipStream_t stream)
{
    const float* x     = (const float*)d_in[0];   // [B,T,C]
    const float* Wqkv  = (const float*)d_in[1];   // [C,3C]
    const float* bqkv  = (const float*)d_in[2];   // [3C]
    const float* Wproj = (const float*)d_in[3];   // [C,C]
    const float* bproj = (const float*)d_in[4];   // [C]
    float* out = (float*)d_out;                   // [B,T,C]

    float* qkv = (float*)d_ws;                         // MROWS * C3 floats (201 MB)
    float* yat = qkv + (size_t)MROWS * C3;             // MROWS * C  floats (67 MB)

    // 1) qkv = x @ Wqkv + bqkv   (M=8192, K=2048, N=6144)
    {
        dim3 grid(MROWS / 128, C3 / 64);
        gemm_bias_wmma_f32<<<grid, 256, 0, stream>>>(x, Wqkv, bqkv, qkv,
                                                     MROWS, C3, Cc);
    }
    // 2) RoPE in place on q,k
    {
        const long total = (long)Bb * Tt * 2 * NH * (HD / 2);  // 16,777,216
        rope_inplace<<<(int)(total / 256), 256, 0, stream>>>(qkv);
    }
    // 3) causal flash attention -> yat [B,T,C]
    {
        dim3 grid(Tt / 64, Bb * NH);
        flash_attn_wmma_f32<<<grid, 128, 0, stream>>>(qkv, yat);
    }
    // 4) out = yat @ Wproj + bproj   (M=8192, K=2048, N=2048)
    {
        dim3 grid(MROWS / 128, Cc / 64);
        gemm_bias_wmma_f32<<<grid, 256, 0, stream>>>(yat, Wproj, bproj, out,
                                                     MROWS, Cc, Cc);
    }
}